// DynamicQuantizer_85993835200580
// MI455X (gfx1250) — compile-verified
//
#include <hip/hip_runtime.h>
#include <math.h>

typedef float v2f __attribute__((ext_vector_type(2)));
typedef float v8f __attribute__((ext_vector_type(8)));

#define EPSF      1e-8f
#define THRESH    0.35f
#define QMIN     -8.0f
#define QMAX      7.0f
#define KSPLIT    8     // K-dim block split for rowpass (latency hiding)
#define RSPLIT    32    // row-dim block split for colpass

// scalars layout in ws: [0]=frob_sq  [1]=sigma_sq  [2]=max|X| bits  [3]=max|X*f| bits

// ---------------------------------------------------------------------------
// prep: f[h] = 1 + 0.5*la*lb; zero v1,v2 (H), u1,u2,w (N), scalar accumulators
// ---------------------------------------------------------------------------
__global__ void prep_kernel(const float* __restrict__ la, const float* __restrict__ lb,
                            float* __restrict__ f, float* __restrict__ v1,
                            float* __restrict__ v2, float* __restrict__ u1,
                            float* __restrict__ u2, float* __restrict__ w,
                            float* __restrict__ scalars, int H, int N) {
  int i = blockIdx.x * blockDim.x + threadIdx.x;
  if (i < H) {
    f[i]  = 1.0f + 0.5f * la[i] * lb[i];
    v1[i] = 0.0f;
    v2[i] = 0.0f;
  }
  if (i < N) {
    u1[i] = 0.0f;
    u2[i] = 0.0f;
    w[i]  = 0.0f;
  }
  if (blockIdx.x == 0 && threadIdx.x < 8) scalars[threadIdx.x] = 0.0f;
}

// ---------------------------------------------------------------------------
// Row pass: partial u = X[:, kchunk] @ v[kchunk] via V_WMMA_F32_16X16X4_F32.
// Grid: (N/128, KSPLIT). 16 rows/wave, kLen columns/block; partial dots are
// atomically accumulated into u_out (linearity of the dot product).
// FROB: also accumulate sum(X^2).  FINAL: also track max|X| and max|X*f|.
// A 16x4 layout: lane<16 -> row=lane, K=k,k+1 ; lane>=16 -> K=k+2,k+3.
// B  4x16 (all columns = v chunk): same per-lane float2 as A.
// D 16x16: lane 0 holds rows 0..7 in c[0..7], lane 16 holds rows 8..15.
// ---------------------------------------------------------------------------
template <bool FROB, bool FINAL>
__global__ void rowpass_kernel(const float* __restrict__ X, const float* __restrict__ v,
                               const float* __restrict__ f, float* __restrict__ u_out,
                               float* __restrict__ scalars, int H, int kLen) {
  __shared__ float red[256];
  const int tid     = threadIdx.x;
  const int lane    = tid & 31;
  const int wave    = tid >> 5;
  const int rowBase = (blockIdx.x * (int)(blockDim.x >> 5) + wave) * 16;
  const int m       = lane & 15;
  const int koff    = (lane >> 4) * 2;
  const int k0      = blockIdx.y * kLen;

  const float* __restrict__ xrow = X + (size_t)(rowBase + m) * (size_t)H + k0 + koff;
  const float* __restrict__ vk   = v + k0 + koff;
  const float* __restrict__ fk   = f + k0 + koff;

  v8f c0 = {}; v8f c1 = {};
  float fr = 0.0f, mx0 = 0.0f, mx1 = 0.0f;

  for (int k = 0; k < kLen; k += 8) {
    v2f a0 = *(const v2f*)(xrow + k);
    v2f b0 = *(const v2f*)(vk + k);
    v2f a1 = *(const v2f*)(xrow + k + 4);
    v2f b1 = *(const v2f*)(vk + k + 4);
#if __has_builtin(__builtin_amdgcn_wmma_f32_16x16x4_f32)
    c0 = __builtin_amdgcn_wmma_f32_16x16x4_f32(false, a0, false, b0, (short)0, c0, false, false);
    c1 = __builtin_amdgcn_wmma_f32_16x16x4_f32(false, a1, false, b1, (short)0, c1, false, false);
#else
    c0[0] += a0.x * b0.x + a0.y * b0.y;   // compile-probe fallback only
    c1[0] += a1.x * b1.x + a1.y * b1.y;
#endif
    if (FROB) fr += a0.x * a0.x + a0.y * a0.y + a1.x * a1.x + a1.y * a1.y;
    if (FINAL) {
      mx0 = fmaxf(mx0, fmaxf(fmaxf(fabsf(a0.x), fabsf(a0.y)),
                             fmaxf(fabsf(a1.x), fabsf(a1.y))));
      v2f f0 = *(const v2f*)(fk + k);
      v2f f1 = *(const v2f*)(fk + k + 4);
      mx1 = fmaxf(mx1, fmaxf(fmaxf(fabsf(a0.x * f0.x), fabsf(a0.y * f0.y)),
                             fmaxf(fabsf(a1.x * f1.x), fabsf(a1.y * f1.y))));
    }
  }
  v8f c = c0 + c1;

  if (m == 0) {  // lanes 0 and 16 own distinct row halves of D
    float* dst = u_out + rowBase + (lane >> 4) * 8;
#pragma unroll
    for (int j = 0; j < 8; ++j) atomicAdd(dst + j, c[j]);
  }

  if (FINAL) {
    red[tid] = mx0; __syncthreads();
    for (int off = 128; off > 0; off >>= 1) {
      if (tid < off) red[tid] = fmaxf(red[tid], red[tid + off]);
      __syncthreads();
    }
    if (tid == 0) atomicMax((unsigned int*)(scalars + 2), __float_as_uint(red[0]));
    __syncthreads();
    red[tid] = mx1; __syncthreads();
    for (int off = 128; off > 0; off >>= 1) {
      if (tid < off) red[tid] = fmaxf(red[tid], red[tid + off]);
      __syncthreads();
    }
    if (tid == 0) atomicMax((unsigned int*)(scalars + 3), __float_as_uint(red[0]));
  }

  if (FROB) {
    red[tid] = fr; __syncthreads();
    for (int off = 128; off > 0; off >>= 1) {
      if (tid < off) red[tid] += red[tid + off];
      __syncthreads();
    }
    if (tid == 0) atomicAdd(scalars + 0, red[0]);
  }
}

// ---------------------------------------------------------------------------
// Column pass: v_out[h] += sum_{n in chunk} X[n,h] * u[n]
// float2 per thread (b64 coalesced loads), row-chunked, atomic accumulate.
// ---------------------------------------------------------------------------
__global__ void colpass_kernel(const float* __restrict__ X, const float* __restrict__ u,
                               float* __restrict__ v_out, int H, int rowsPerBlock) {
  const int h2 = (blockIdx.x * blockDim.x + threadIdx.x) * 2;
  const int n0 = blockIdx.y * rowsPerBlock;
  const float* __restrict__ xp = X + (size_t)n0 * (size_t)H + h2;
  const float* __restrict__ up = u + n0;
  v2f acc = {0.0f, 0.0f};
  for (int n = 0; n < rowsPerBlock; n += 4) {
    float u0 = up[n], u1 = up[n + 1], u2 = up[n + 2], u3 = up[n + 3];
    v2f x0 = *(const v2f*)(xp + (size_t)(n + 0) * H);
    v2f x1 = *(const v2f*)(xp + (size_t)(n + 1) * H);
    v2f x2 = *(const v2f*)(xp + (size_t)(n + 2) * H);
    v2f x3 = *(const v2f*)(xp + (size_t)(n + 3) * H);
    acc.x = fmaf(x0.x, u0, acc.x); acc.y = fmaf(x0.y, u0, acc.y);
    acc.x = fmaf(x1.x, u1, acc.x); acc.y = fmaf(x1.y, u1, acc.y);
    acc.x = fmaf(x2.x, u2, acc.x); acc.y = fmaf(x2.y, u2, acc.y);
    acc.x = fmaf(x3.x, u3, acc.x); acc.y = fmaf(x3.y, u3, acc.y);
  }
  atomicAdd(v_out + h2, acc.x);
  atomicAdd(v_out + h2 + 1, acc.y);
}

// ---------------------------------------------------------------------------
// In-place normalize (single block): v /= max(||v||, eps)
// ---------------------------------------------------------------------------
__global__ void normalize_kernel(float* __restrict__ v, int n) {
  __shared__ float red[1024];
  __shared__ float s_inv;
  float s = 0.0f;
  for (int i = threadIdx.x; i < n; i += blockDim.x) { float x = v[i]; s += x * x; }
  red[threadIdx.x] = s; __syncthreads();
  for (int off = blockDim.x >> 1; off > 0; off >>= 1) {
    if ((int)threadIdx.x < off) red[threadIdx.x] += red[threadIdx.x + off];
    __syncthreads();
  }
  if (threadIdx.x == 0) s_inv = 1.0f / fmaxf(sqrtf(red[0]), EPSF);
  __syncthreads();
  float inv = s_inv;
  for (int i = threadIdx.x; i < n; i += blockDim.x) v[i] *= inv;
}

// ---------------------------------------------------------------------------
// sigma_sq += sum(w^2)   (w = X @ v_final, accumulated by the final rowpass)
// ---------------------------------------------------------------------------
__global__ void sumsq_kernel(const float* __restrict__ w, float* __restrict__ scalars, int n) {
  __shared__ float red[256];
  float s = 0.0f;
  for (int i = blockIdx.x * blockDim.x + threadIdx.x; i < n; i += gridDim.x * blockDim.x) {
    float x = w[i]; s += x * x;
  }
  red[threadIdx.x] = s; __syncthreads();
  for (int off = 128; off > 0; off >>= 1) {
    if ((int)threadIdx.x < off) red[threadIdx.x] += red[threadIdx.x + off];
    __syncthreads();
  }
  if (threadIdx.x == 0) atomicAdd(scalars + 1, red[0]);
}

// ---------------------------------------------------------------------------
// Quantize: out = clamp(rint(X_adapt/scale), -8, 7) * scale   (float4 path)
// ---------------------------------------------------------------------------
__global__ void quant_kernel(const float* __restrict__ X, const float* __restrict__ f,
                             const float* __restrict__ scalars, float* __restrict__ out,
                             int H, long long total4) {
  long long i = (long long)blockIdx.x * blockDim.x + threadIdx.x;
  if (i >= total4) return;

  const float frob  = scalars[0];
  const float sig   = scalars[1];
  const bool  cond  = (sig / (frob + EPSF)) > THRESH;
  const float maxv  = __uint_as_float(((const unsigned int*)scalars)[cond ? 3 : 2]);
  const float scale = (maxv < EPSF) ? 1.0f : maxv * (1.0f / 7.0f);
  const float invs  = 1.0f / scale;

  size_t base = (size_t)i * 4;
  int    col  = (int)(base % (size_t)H);  // H % 4 == 0, no wrap inside float4
  float4 x    = *(const float4*)(X + base);
  float4 fc   = cond ? *(const float4*)(f + col) : make_float4(1.f, 1.f, 1.f, 1.f);

  float4 o;
  o.x = fminf(fmaxf(rintf(x.x * fc.x * invs), QMIN), QMAX) * scale;
  o.y = fminf(fmaxf(rintf(x.y * fc.y * invs), QMIN), QMAX) * scale;
  o.z = fminf(fmaxf(rintf(x.z * fc.z * invs), QMIN), QMAX) * scale;
  o.w = fminf(fmaxf(rintf(x.w * fc.w * invs), QMIN), QMAX) * scale;
  *(float4*)(out + base) = o;
}

// ---------------------------------------------------------------------------
extern "C" void kernel_launch(void* const* d_in, const int* in_sizes, int n_in,
                              void* d_out, int out_size, void* d_ws, size_t ws_size,
                              hipStream_t stream) {
  const float* X  = (const float*)d_in[0];
  const float* la = (const float*)d_in[1];
  const float* lb = (const float*)d_in[2];
  const float* v0 = (const float*)d_in[3];

  const int H = in_sizes[1];           // 4096
  const int N = in_sizes[0] / H;       // 8192

  float* wsf     = (float*)d_ws;
  float* f       = wsf;                 // [H]
  float* v1      = wsf + H;             // [H]
  float* v2      = wsf + 2 * H;         // [H]
  float* u1      = wsf + 3 * H;         // [N]
  float* u2      = u1 + N;              // [N]
  float* w       = u2 + N;              // [N]
  float* scalars = w + N;               // [8]

  const dim3 rowGrid(N / 128, KSPLIT);  // 8 waves/block * 16 rows/wave
  const int  kLen = H / KSPLIT;
  const dim3 colGrid(H / 512, RSPLIT);  // 2 cols/thread
  const int  rowsPerColBlock = N / RSPLIT;
  const int  maxHN = (N > H ? N : H);

  // 0) per-column factor + zero all accumulators (every call: deterministic)
  prep_kernel<<<(maxHN + 255) / 256, 256, 0, stream>>>(la, lb, f, v1, v2, u1, u2, w,
                                                       scalars, H, N);

  // power iteration with deferred normalization (direction-preserving)
  // 1) u1 = X v0  (+ frob_sq)
  rowpass_kernel<true, false><<<rowGrid, 256, 0, stream>>>(X, v0, f, u1, scalars, H, kLen);
  // 2) v1 = X^T u1 ; normalize
  colpass_kernel<<<colGrid, 256, 0, stream>>>(X, u1, v1, H, rowsPerColBlock);
  normalize_kernel<<<1, 1024, 0, stream>>>(v1, H);
  // 3) u2 = X v1
  rowpass_kernel<false, false><<<rowGrid, 256, 0, stream>>>(X, v1, f, u2, scalars, H, kLen);
  // 4) v2 = X^T u2 ; normalize
  colpass_kernel<<<colGrid, 256, 0, stream>>>(X, u2, v2, H, rowsPerColBlock);
  normalize_kernel<<<1, 1024, 0, stream>>>(v2, H);
  // 5) w = X v2 (partials), fused max|X| and max|X*f| ; then sigma^2 = ||w||^2
  rowpass_kernel<false, true><<<rowGrid, 256, 0, stream>>>(X, v2, f, w, scalars, H, kLen);
  sumsq_kernel<<<8, 256, 0, stream>>>(w, scalars, N);
  // 6) fake-quantize
  long long total4 = (long long)N * H / 4;
  quant_kernel<<<(unsigned)((total4 + 255) / 256), 256, 0, stream>>>(
      X, f, scalars, (float*)d_out, H, total4);
}